// SWD_17205638988371
// MI455X (gfx1250) — compile-verified
//
#include <hip/hip_runtime.h>
#include <stdint.h>

// Problem constants (from reference setup_inputs: B=8, L=4096, D=1024, fp32)
#define B_      8
#define L_      4096
#define D_      1024
#define TC      32          // columns per tile
#define TR      256         // output rows per tile (even)
#define PAIRS   (TR / 2)    // 128 row-pairs per tile
#define NB      320         // LDS band rows: TR + 2*(TC-1)=318, rounded to 320
#define THREADS 256

// ---- CDNA5 async global->LDS path (gfx1250), with safe fallback -------------
#if defined(__HIP_DEVICE_COMPILE__) && __has_builtin(__builtin_amdgcn_global_load_async_to_lds_b128)
#define USE_ASYNC_LDS 1
#else
#define USE_ASYNC_LDS 0
#endif

#if USE_ASYNC_LDS
// Builtin signature (from hipcc diagnostic): param 1 is
//   '__attribute__((__vector_size__(4 * sizeof(int)))) int __device__ *'
// i.e. a GCC-style int4 pointer in the global (AS1, printed "__device__")
// address space. Param 2 is the LDS-side (AS3) counterpart.
typedef int v4i_t __attribute__((__vector_size__(4 * sizeof(int))));
typedef __attribute__((address_space(1))) v4i_t gv4i_t;   // global int4
typedef __attribute__((address_space(3))) v4i_t lv4i_t;   // LDS int4

static __device__ __forceinline__ void async_copy_b128(const float* gp, float* lp) {
    // (global src, LDS dst, imm offset, imm cpol)
    __builtin_amdgcn_global_load_async_to_lds_b128(
        (gv4i_t*)(uintptr_t)gp,
        (lv4i_t*)(uint32_t)(uintptr_t)lp,
        0, 0);
}

static __device__ __forceinline__ void wait_async_lds() {
#if __has_builtin(__builtin_amdgcn_s_wait_asynccnt)
    __builtin_amdgcn_s_wait_asynccnt(0);
#else
    asm volatile("s_wait_asynccnt 0" ::: "memory");
#endif
}
#endif
// -----------------------------------------------------------------------------

__global__ __launch_bounds__(THREADS)
void butterfly_pairsort_kernel(const float* __restrict__ v, float* __restrict__ out)
{
    // Band of the input, row-major [NB][TC]. 40 KB -> 8 workgroups / 320KB WGP.
    __shared__ float lds[NB * TC];

    const int tid = threadIdx.x;
    const int rt  = blockIdx.x;            // row tile   (fastest -> L2 reuse of band overlap)
    const int jt  = blockIdx.y;            // column tile
    const int b   = blockIdx.z;            // batch

    const int j0   = jt * TC;
    const int r0   = rt * TR;
    // Max shift over this column tile: s_j = 2*(j-1) for j>=2, 0 for j<2.
    // max over j in [j0, j0+TC) is 2*(j0+TC-2)  (also correct for j0==0).
    const int sMax = 2 * (j0 + TC - 2);
    const int bandStart = (r0 - sMax + L_) & (L_ - 1);    // sMax < L, so one +L suffices

    const size_t baseIn = (size_t)b * L_ * D_;

    // ---------------- Load diagonal band into LDS (coalesced b128) ----------
    // NB*TC floats = 2560 float4 chunks; THREADS=256 -> exactly 10 iterations.
    #pragma unroll
    for (int it = 0; it < (NB * TC / 4) / THREADS; ++it) {
        const int linear = it * THREADS + tid;
        const int lr = linear >> 3;              // band row     [0, NB)
        const int c4 = (linear & 7) << 2;        // column *4    {0,4,...,28}
        const int gi = (bandStart + lr) & (L_ - 1);
        const float* gp = v + baseIn + (size_t)gi * D_ + (j0 + c4);
        float*       lp = &lds[lr * TC + c4];
#if USE_ASYNC_LDS
        async_copy_b128(gp, lp);                 // GLOBAL_LOAD_ASYNC_TO_LDS_B128
#else
        *(float4*)lp = *(const float4*)gp;       // global_load_b128 + ds_store_b128
#endif
    }
#if USE_ASYNC_LDS
    wait_async_lds();                            // own ASYNCcnt before barrier signal
#endif
    __syncthreads();

    // ---------------- Pairwise min/max + coalesced NT stores ----------------
    const int c = tid & 31;                      // lane == column within tile
    const int w = tid >> 5;                      // wave id 0..7
    const int j = j0 + c;
    const int s = (j < 2) ? 0 : 2 * (j - 1);     // < L_, no mod needed
    const int d = sMax - s;                      // band-row offset for this column

    // TC*PAIRS = 4096 pair tasks / 256 threads = 16 iterations.
    #pragma unroll 4
    for (int it = 0; it < PAIRS / 8; ++it) {
        const int kk  = it * 8 + w;              // pair index 0..127
        const int lrr = 2 * kk + d;              // <= 318; lrr+1 <= 319 < NB
        // LDS bank stride across lanes: -63 words == 1 mod 64 -> conflict-free
        const float a0 = lds[lrr * TC + c];
        const float a1 = lds[(lrr + 1) * TC + c];
        const float mn = fminf(a0, a1);
        const float mx = fmaxf(a0, a1);
        const size_t o = baseIn + (size_t)(r0 + 2 * kk) * D_ + j;
        // Output is never re-read: non-temporal, keep L2 for the input band.
        __builtin_nontemporal_store(mn, out + o);        // row 2k   : min
        __builtin_nontemporal_store(mx, out + o + D_);   // row 2k+1 : max
    }
}

extern "C" void kernel_launch(void* const* d_in, const int* in_sizes, int n_in,
                              void* d_out, int out_size, void* d_ws, size_t ws_size,
                              hipStream_t stream) {
    const float* v   = (const float*)d_in[0];
    float*       out = (float*)d_out;
    (void)in_sizes; (void)n_in; (void)out_size; (void)d_ws; (void)ws_size;

    dim3 grid(L_ / TR, D_ / TC, B_);   // (16, 32, 8) = 4096 blocks
    butterfly_pairsort_kernel<<<grid, THREADS, 0, stream>>>(v, out);
}